// XConvLayerCoreV1_86277303042223
// MI455X (gfx1250) — compile-verified
//
#include <hip/hip_runtime.h>
#include <cmath>

// Problem constants (from reference)
#define B_   8
#define N_   4096
#define F_   64
#define P_   1024
#define K_   16
#define D_   2
#define C_   128
#define CPF_ 32
#define CF_  96          // CPF + F
#define EPS_ 1e-3f

typedef __attribute__((ext_vector_type(2))) float v2f;
typedef __attribute__((ext_vector_type(8))) float v8f;
typedef __attribute__((ext_vector_type(4))) unsigned int u32x4;
typedef __attribute__((ext_vector_type(8))) int i32x8;
typedef __attribute__((ext_vector_type(4))) int i32x4;
// Payload type for the async-to-LDS b128 builtin (matches clang's expected
// parameter type: pointer to 16-byte int vector in AS1/AS3).
typedef int vb128_t __attribute__((__vector_size__(16)));

// Feature detection (device pass only; host pass falls back harmlessly)
#if __has_builtin(__builtin_amdgcn_global_load_async_to_lds_b128)
#define HAVE_ASYNC_GATHER 1
#endif
#if __has_builtin(__builtin_amdgcn_tensor_load_to_lds)
#define HAVE_TDM 1
#endif

__device__ __forceinline__ float eluf(float x) { return x > 0.0f ? x : expm1f(x); }

// bn params stacked [gamma, beta, mean, var] each of length nch
__device__ __forceinline__ float bnf(const float* __restrict__ bn, int nch, int ch, float x) {
  float g = bn[ch], b = bn[nch + ch], m = bn[2 * nch + ch], v = bn[3 * nch + ch];
  return (x - m) * g * rsqrtf(v + EPS_) + b;
}

// ---------------------------------------------------------------------------
// Kernel 1: KNN (top-32 nearest, dilation 2 -> 16 indices) + qrs copy.
// One 256-thread block per query point. Distance row lives in LDS; 32 rounds
// of block argmin with lowest-index tie-break (matches lax.top_k ordering).
// ---------------------------------------------------------------------------
__global__ __launch_bounds__(256) void knn_kernel(const float* __restrict__ pts,
                                                  int* __restrict__ nn_idx,
                                                  float* __restrict__ qrs_out) {
  __shared__ float dist[N_];
  __shared__ float rv[256];
  __shared__ int   ri[256];
  __shared__ int   chosen[2 * K_];

  const int bp  = blockIdx.x;
  const int b   = bp / P_;
  const int p   = bp % P_;
  const int tid = threadIdx.x;

  const float qx = pts[(b * N_ + p) * 3 + 0];
  const float qy = pts[(b * N_ + p) * 3 + 1];
  const float qz = pts[(b * N_ + p) * 3 + 2];
  const float q2 = qx * qx + qy * qy + qz * qz;

  for (int n = tid; n < N_; n += 256) {
    float px = pts[(b * N_ + n) * 3 + 0];
    float py = pts[(b * N_ + n) * 3 + 1];
    float pz = pts[(b * N_ + n) * 3 + 2];
    float dt = qx * px + qy * py + qz * pz;
    float p2 = px * px + py * py + pz * pz;
    dist[n] = q2 - 2.0f * dt + p2;   // same expanded form as the reference
  }
  __syncthreads();

  for (int r = 0; r < 2 * K_; ++r) {
    float bv = __builtin_inff();
    int   bi = 0x7fffffff;
    for (int n = tid; n < N_; n += 256) {
      float v = dist[n];
      if (v < bv || (v == bv && n < bi)) { bv = v; bi = n; }
    }
    rv[tid] = bv; ri[tid] = bi;
    __syncthreads();
    for (int s = 128; s > 0; s >>= 1) {
      if (tid < s) {
        float ov = rv[tid + s]; int oi = ri[tid + s];
        if (ov < rv[tid] || (ov == rv[tid] && oi < ri[tid])) { rv[tid] = ov; ri[tid] = oi; }
      }
      __syncthreads();
    }
    if (tid == 0) { chosen[r] = ri[0]; dist[ri[0]] = __builtin_inff(); }
    __syncthreads();
  }

  if (tid < K_) nn_idx[bp * K_ + tid] = chosen[tid * D_];   // idx[:, :, ::2]
  if (tid < 3)  qrs_out[bp * 3 + tid] = pts[(b * N_ + p) * 3 + tid];
}

// ---------------------------------------------------------------------------
// Kernel 2: X-Conv core. One wave32 per point; 2 waves per 64-thread block.
// WMMA f32 16x16x4 for: local@w_fts0, f@w_fts1, and fts_X = x2 @ nn_in.
// Feature gather uses GLOBAL_LOAD_ASYNC_TO_LDS; w_fts1 staged via TDM.
// ---------------------------------------------------------------------------
#define WAVES_PER_BLOCK 2
#define SM_PER_WAVE 5184
#define OF_LOCAL 0      // 16x4  (64)   local coords, col 3 zero-padded
#define OF_F     64     // 16x32 (512)  f intermediate; reused later as tS
#define OF_NNIN  576    // 16x96 (1536) [f1 | gathered features]
#define OF_XA    2112   // 256
#define OF_XB    2368   // 256
#define OF_FTSX  2624   // 16x96 (1536)
#define OF_W1    4160   // 32x32 (1024) TDM-staged w_fts1

#ifdef HAVE_TDM
// Stage 1024 contiguous f32 (w_fts1) from global into LDS via the Tensor Data
// Mover. 1-row descriptor: data_size=4B, tensor_dim0=tile_dim0=1024,
// tensor_dim1=tile_dim1=1, dim0_stride=1024, type=2, count=1.
__device__ __forceinline__ void tdm_stage_w1(const float* gsrc, float* ldst) {
  unsigned long long ga = (unsigned long long)(uintptr_t)gsrc;
  unsigned int lo = (unsigned int)(uintptr_t)ldst;  // low 32 bits == LDS byte offset
  u32x4 g0 = {};
  g0[0] = 1u;                                        // count=1, is_restore=0
  g0[1] = lo;                                        // lds_addr
  g0[2] = (unsigned int)(ga & 0xffffffffu);          // global_addr[31:0]
  g0[3] = (unsigned int)((ga >> 32) & 0x01ffffffu)   // global_addr[56:32]
        | (2u << 30);                                // type=2 ("image")
  i32x8 g1 = {};
  g1[0] = (int)(2u << 16);                           // data_size=4B
  g1[1] = (int)(0x400u << 16);                       // tensor_dim0[15:0]=1024 @bits63:48
  g1[2] = (int)(1u << 16);                           // tensor_dim0[31:16]=0 | tensor_dim1=1
  g1[3] = (int)(0x400u << 16);                       // tile_dim0=1024 @bits127:112
  g1[4] = 1;                                         // tile_dim1=1, tile_dim2=0
  g1[5] = 1024;                                      // tensor_dim0_stride[31:0]
  g1[6] = 0; g1[7] = 0;
  i32x4 g2 = {};
  i32x4 g3 = {};
#if __clang_major__ >= 23
  i32x8 g4 = {};
  __builtin_amdgcn_tensor_load_to_lds(g0, g1, g2, g3, g4, 0);
#else
  __builtin_amdgcn_tensor_load_to_lds(g0, g1, g2, g3, 0);
#endif
}
#endif

__global__ __launch_bounds__(64) void xconv_kernel(
    const float* __restrict__ pts,    const float* __restrict__ feats,
    const float* __restrict__ w_fts0, const float* __restrict__ bn_fts0,
    const float* __restrict__ w_fts1, const float* __restrict__ bn_fts1,
    const float* __restrict__ w_x0,   const float* __restrict__ bn_x0,
    const float* __restrict__ w_x1,   const float* __restrict__ bn_x1,
    const float* __restrict__ w_x2,   const float* __restrict__ bn_x2,
    const float* __restrict__ w_dw,   const float* __restrict__ w_pw,
    const float* __restrict__ bn_sep,
    const int* __restrict__ nn_idx,   float* __restrict__ out) {
  __shared__ float smem[WAVES_PER_BLOCK][SM_PER_WAVE];
  __shared__ int   sidx[WAVES_PER_BLOCK][K_];

  const int wave = threadIdx.x >> 5;
  const int lane = threadIdx.x & 31;
  const int half = lane >> 4;        // which 16-lane half
  const int m16  = lane & 15;
  const int bp   = blockIdx.x * WAVES_PER_BLOCK + wave;
  const int b    = bp / P_;
  const int p    = bp % P_;

  float* localS = smem[wave] + OF_LOCAL;
  float* fS     = smem[wave] + OF_F;
  float* nnin   = smem[wave] + OF_NNIN;
  float* xA     = smem[wave] + OF_XA;
  float* xB     = smem[wave] + OF_XB;
  float* ftsX   = smem[wave] + OF_FTSX;
  float* w1S    = smem[wave] + OF_W1;
  float* tS     = fS;  // fS dead after f1 stage; reuse for depthwise temps

  // Warm the big weight tables into cache (global_prefetch_b8).
  __builtin_prefetch(w_pw + (size_t)threadIdx.x * 384, 0, 0);   // 96*2*128 floats
  __builtin_prefetch(w_x0 + (size_t)threadIdx.x * 192, 0, 0);   // 16*3*256 floats

#ifdef HAVE_TDM
  // Kick off the TDM copy of w_fts1 (4KB) into this wave's LDS staging buffer.
  tdm_stage_w1(w_fts1, w1S);
#endif

  const float qx = pts[(b * N_ + p) * 3 + 0];
  const float qy = pts[(b * N_ + p) * 3 + 1];
  const float qz = pts[(b * N_ + p) * 3 + 2];

  if (lane < K_) {
    int ik = nn_idx[bp * K_ + lane];
    sidx[wave][lane] = ik;
    localS[lane * 4 + 0] = pts[(b * N_ + ik) * 3 + 0] - qx;
    localS[lane * 4 + 1] = pts[(b * N_ + ik) * 3 + 1] - qy;
    localS[lane * 4 + 2] = pts[(b * N_ + ik) * 3 + 2] - qz;
    localS[lane * 4 + 3] = 0.0f;     // pad K=3 -> 4 for WMMA
  }
  __syncthreads();

  // Gather neighbor features into nnin[:, 32:96] (16 rows x 256B each).
#ifdef HAVE_ASYNC_GATHER
  {
    const int rhalf = lane >> 4;          // 2 rows per instruction
    const int f4    = (lane & 15) * 4;    // 4 floats (16B) per lane
    for (int i = 0; i < 8; ++i) {
      int k = i * 2 + rhalf;
      const float* gp = feats + (size_t)(b * N_ + sidx[wave][k]) * F_ + f4;
      float*       lp = nnin + k * CF_ + CPF_ + f4;
      __builtin_amdgcn_global_load_async_to_lds_b128(
          (vb128_t __attribute__((address_space(1)))*)(uintptr_t)gp,
          (vb128_t __attribute__((address_space(3)))*)(unsigned int)(uintptr_t)lp,
          0, 0);
    }
#if __has_builtin(__builtin_amdgcn_s_wait_asynccnt)
    __builtin_amdgcn_s_wait_asynccnt(0);
#else
    asm volatile("s_wait_asynccnt 0" ::: "memory");
#endif
  }
#else
  for (int e = lane; e < K_ * F_; e += 32) {
    int k = e >> 6;
    int f = e & 63;
    nnin[k * CF_ + CPF_ + f] = feats[(size_t)(b * N_ + sidx[wave][k]) * F_ + f];
  }
#endif

  // ---- f = bn(elu(local @ w_fts0)) : 16x32, two 16x16 WMMA tiles, K=4 ----
  for (int t = 0; t < 2; ++t) {
    int col = t * 16 + m16;
    v2f a, bb;
    a.x = localS[m16 * 4 + 2 * half + 0];
    a.y = localS[m16 * 4 + 2 * half + 1];
    bb.x = (2 * half + 0 < 3) ? w_fts0[(2 * half + 0) * CPF_ + col] : 0.0f;
    bb.y = (2 * half + 1 < 3) ? w_fts0[(2 * half + 1) * CPF_ + col] : 0.0f;
    v8f acc = {};
    acc = __builtin_amdgcn_wmma_f32_16x16x4_f32(false, a, false, bb, (short)0, acc, false, false);
    for (int r = 0; r < 8; ++r)
      fS[(r + 8 * half) * CPF_ + col] = bnf(bn_fts0, CPF_, col, eluf(acc[r]));
  }
  __syncthreads();

#ifdef HAVE_TDM
  __builtin_amdgcn_s_wait_tensorcnt(0);   // w1S ready
  const float* w1p = w1S;                 // LDS-resident weights
#else
  const float* w1p = w_fts1;
#endif

  // ---- f1 = bn(elu(f @ w_fts1)) -> nnin[:, 0:32] : 2 tiles x 8 K-chunks ----
  for (int t = 0; t < 2; ++t) {
    int col = t * 16 + m16;
    v8f acc = {};
    for (int kc = 0; kc < 8; ++kc) {
      int k0 = kc * 4 + 2 * half;
      v2f a, bb;
      a.x = fS[m16 * CPF_ + k0 + 0];
      a.y = fS[m16 * CPF_ + k0 + 1];
      bb.x = w1p[(k0 + 0) * CPF_ + col];
      bb.y = w1p[(k0 + 1) * CPF_ + col];
      acc = __builtin_amdgcn_wmma_f32_16x16x4_f32(false, a, false, bb, (short)0, acc, false, false);
    }
    for (int r = 0; r < 8; ++r)
      nnin[(r + 8 * half) * CF_ + col] = bnf(bn_fts1, CPF_, col, eluf(acc[r]));
  }

  // ---- x0 = bn(elu(local(48) @ w_x0(48x256))) -> xA[256] (lane-parallel) ----
  for (int t = 0; t < 8; ++t) {
    int o = lane + 32 * t;
    float s = 0.0f;
    for (int k = 0; k < K_; ++k) {
      const float* w = w_x0 + (k * 3) * 256 + o;
      s += localS[k * 4 + 0] * w[0] + localS[k * 4 + 1] * w[256] + localS[k * 4 + 2] * w[512];
    }
    xA[o] = bnf(bn_x0, 256, o, eluf(s));  // xA[k*16+c] layout
  }
  __syncthreads();

  // ---- x1[c][m] = bn(elu(sum_k x0[k][c]*w_x1[k][c][m])) -> xB ----
  for (int t = 0; t < 8; ++t) {
    int o = lane + 32 * t;
    int c = o >> 4, mm = o & 15;
    float s = 0.0f;
    for (int k = 0; k < K_; ++k)
      s += xA[k * 16 + c] * w_x1[(k * 16 + c) * 16 + mm];
    xB[o] = bnf(bn_x1, 256, o, eluf(s));
  }
  __syncthreads();

  // ---- x2[c][m] = bn(sum_k x1[k][c]*w_x2[k][c][m]) (no elu) -> xA ----
  for (int t = 0; t < 8; ++t) {
    int o = lane + 32 * t;
    int c = o >> 4, mm = o & 15;
    float s = 0.0f;
    for (int k = 0; k < K_; ++k)
      s += xB[k * 16 + c] * w_x2[(k * 16 + c) * 16 + mm];
    xA[o] = bnf(bn_x2, 256, o, s);
  }
  __syncthreads();

  // ---- fts_X(16x96) = x2(16x16) @ nnin(16x96): 6 tiles x 4 K-chunks WMMA ----
  for (int t = 0; t < 6; ++t) {
    v8f acc = {};
    for (int kc = 0; kc < 4; ++kc) {
      int k0 = kc * 4 + 2 * half;
      v2f a, bb;
      a.x = xA[m16 * 16 + k0 + 0];
      a.y = xA[m16 * 16 + k0 + 1];
      bb.x = nnin[(k0 + 0) * CF_ + t * 16 + m16];
      bb.y = nnin[(k0 + 1) * CF_ + t * 16 + m16];
      acc = __builtin_amdgcn_wmma_f32_16x16x4_f32(false, a, false, bb, (short)0, acc, false, false);
    }
    for (int r = 0; r < 8; ++r)
      ftsX[(r + 8 * half) * CF_ + t * 16 + m16] = acc[r];
  }
  __syncthreads();

  // ---- depthwise: t[c][m] = sum_k ftsX[k][c] * w_dw[k][c][m] ----
  for (int e = lane; e < CF_ * 2; e += 32) {
    int c = e >> 1, mm = e & 1;
    float s = 0.0f;
    for (int k = 0; k < K_; ++k)
      s += ftsX[k * CF_ + c] * w_dw[(k * CF_ + c) * 2 + mm];
    tS[e] = s;
  }
  __syncthreads();

  // ---- pointwise -> bn(elu(.)) -> out[bp, 0:128] ----
  for (int t = 0; t < 4; ++t) {
    int o = lane + 32 * t;
    float s = 0.0f;
    for (int c = 0; c < CF_; ++c) {
      s += tS[2 * c + 0] * w_pw[(c * 2 + 0) * C_ + o];
      s += tS[2 * c + 1] * w_pw[(c * 2 + 1) * C_ + o];
    }
    out[(size_t)bp * C_ + o] = bnf(bn_sep, C_, o, eluf(s));
  }
}

extern "C" void kernel_launch(void* const* d_in, const int* in_sizes, int n_in,
                              void* d_out, int out_size, void* d_ws, size_t ws_size,
                              hipStream_t stream) {
  const float* pts     = (const float*)d_in[0];
  const float* feats   = (const float*)d_in[1];
  const float* w_fts0  = (const float*)d_in[2];
  const float* bn_fts0 = (const float*)d_in[3];
  const float* w_fts1  = (const float*)d_in[4];
  const float* bn_fts1 = (const float*)d_in[5];
  const float* w_x0    = (const float*)d_in[6];
  const float* bn_x0   = (const float*)d_in[7];
  const float* w_x1    = (const float*)d_in[8];
  const float* bn_x1   = (const float*)d_in[9];
  const float* w_x2    = (const float*)d_in[10];
  const float* bn_x2   = (const float*)d_in[11];
  const float* w_dw    = (const float*)d_in[12];
  const float* w_pw    = (const float*)d_in[13];
  const float* bn_sep  = (const float*)d_in[14];

  float* qrs_out = (float*)d_out;                        // (B,P,3)
  float* out     = (float*)d_out + (size_t)B_ * P_ * 3;  // (B,P,C)
  int*   nn_idx  = (int*)d_ws;                           // (B*P, K) indices

  knn_kernel<<<B_ * P_, 256, 0, stream>>>(pts, nn_idx, qrs_out);
  xconv_kernel<<<(B_ * P_) / WAVES_PER_BLOCK, 64, 0, stream>>>(
      pts, feats, w_fts0, bn_fts0, w_fts1, bn_fts1,
      w_x0, bn_x0, w_x1, bn_x1, w_x2, bn_x2, w_dw, w_pw, bn_sep,
      nn_idx, out);
}